// CustomBatchNorm2D_72859825209967
// MI455X (gfx1250) — compile-verified
//
#include <hip/hip_runtime.h>
#include <math.h>

typedef __attribute__((ext_vector_type(2))) float v2f;
typedef __attribute__((ext_vector_type(8))) float v8f;

#define N_  32
#define C_  512
#define HW_ 1024   // H*W = 32*32

// -----------------------------------------------------------------------------
// Pass 1: per-(n,c) row sums via V_WMMA_F32_16X16X4_F32 with B = ones.
// One workgroup (8 waves) handles 16 consecutive rows (rows = n*C + c).
// Wave w reduces K-slice [w*128, w*128+128) with 32 chained WMMAs
// (D = A x ones + C accumulates the row sums in f32).
// A layout (32-bit A 16x4): lane l -> row (l&15), K offset ((l>>4)*2, +1).
// D layout (32-bit 16x16): lane 0 holds rows 0..7 in d[0..7]; lane 16 rows 8..15.
// -----------------------------------------------------------------------------
__global__ __launch_bounds__(256)
void bn_rowsum_wmma(const float* __restrict__ x, float* __restrict__ S) {
  __shared__ float partials[8 * 16];

  const int lane    = threadIdx.x & 31;
  const int wave    = threadIdx.x >> 5;
  const int rowBase = blockIdx.x * 16;
  const int m       = lane & 15;         // row within 16-row tile
  const int koff    = (lane >> 4) * 2;   // K sub-offset per A layout

  const float* p = x + (size_t)(rowBase + m) * HW_ + wave * 128 + koff;

  v2f ones; ones[0] = 1.0f; ones[1] = 1.0f;
  v8f acc = {};

#pragma unroll 4
  for (int i = 0; i < 32; ++i) {
    v2f a = *(const v2f*)(p + 4 * i);    // 8B aligned, global_load_b64
    // 8 args: (neg_a, A, neg_b, B, c_mod, C, reuse_a, reuse_b)
    acc = __builtin_amdgcn_wmma_f32_16x16x4_f32(
        false, a, false, ones, (short)0, acc, false, false);
  }

  if (lane == 0) {
#pragma unroll
    for (int j = 0; j < 8; ++j) partials[wave * 16 + j] = acc[j];
  } else if (lane == 16) {
#pragma unroll
    for (int j = 0; j < 8; ++j) partials[wave * 16 + 8 + j] = acc[j];
  }
  __syncthreads();

  if (threadIdx.x < 16) {
    float s = 0.0f;
#pragma unroll
    for (int w = 0; w < 8; ++w) s += partials[w * 16 + threadIdx.x];
    S[rowBase + threadIdx.x] = s;
  }
}

// -----------------------------------------------------------------------------
// Pass 2 (tiny): fold S[n,c] into fused per-channel affine coefficients.
//   mean[c]  = sum_n S / (N*HW)
//   diag[c]  = sum_n (S - HW*mean)^2 / HW
//   A[c]     = gamma[c] * |diag[c]|
//   B[c]     = beta[c]  - A[c] * mean[c]
// -----------------------------------------------------------------------------
__global__ __launch_bounds__(C_)
void bn_stats(const float* __restrict__ S,
              const float* __restrict__ gamma,
              const float* __restrict__ beta,
              float* __restrict__ Acoef,
              float* __restrict__ Bcoef) {
  const int c = threadIdx.x;
  float sv[N_];
  float sum = 0.0f;
#pragma unroll
  for (int n = 0; n < N_; ++n) {
    sv[n] = S[n * C_ + c];
    sum  += sv[n];
  }
  const float mean = sum * (1.0f / ((float)N_ * (float)HW_));
  float diag = 0.0f;
#pragma unroll
  for (int n = 0; n < N_; ++n) {
    float t = sv[n] - (float)HW_ * mean;
    diag = fmaf(t, t, diag);
  }
  diag *= (1.0f / (float)HW_);
  const float scale = fabsf(diag);
  const float a = gamma[c] * scale;
  Acoef[c] = a;
  Bcoef[c] = fmaf(-a, mean, beta[c]);
}

// -----------------------------------------------------------------------------
// Pass 3: out = A[c]*x + B[c], float4-vectorized (x should largely hit L2).
// 256 float4 per (n,c) row -> c = (i4 >> 8) & (C-1).
// -----------------------------------------------------------------------------
__global__ __launch_bounds__(256)
void bn_apply(const float* __restrict__ x,
              const float* __restrict__ Acoef,
              const float* __restrict__ Bcoef,
              float* __restrict__ out) {
  const size_t i4 = (size_t)blockIdx.x * blockDim.x + threadIdx.x;
  const int c = (int)((i4 >> 8) & (C_ - 1));
  const float a = Acoef[c];
  const float b = Bcoef[c];
  float4 v = ((const float4*)x)[i4];
  float4 r;
  r.x = fmaf(a, v.x, b);
  r.y = fmaf(a, v.y, b);
  r.z = fmaf(a, v.z, b);
  r.w = fmaf(a, v.w, b);
  ((float4*)out)[i4] = r;
}

// -----------------------------------------------------------------------------
extern "C" void kernel_launch(void* const* d_in, const int* in_sizes, int n_in,
                              void* d_out, int out_size, void* d_ws, size_t ws_size,
                              hipStream_t stream) {
  const float* x     = (const float*)d_in[0];
  const float* gamma = (const float*)d_in[1];
  const float* beta  = (const float*)d_in[2];
  float* out = (float*)d_out;

  // Workspace layout: S[N*C] | Acoef[C] | Bcoef[C]  (69,632 bytes total)
  float* S     = (float*)d_ws;
  float* Acoef = S + (size_t)N_ * C_;
  float* Bcoef = Acoef + C_;

  const int nRows = N_ * C_;                 // 16384 rows of length HW
  bn_rowsum_wmma<<<nRows / 16, 256, 0, stream>>>(x, S);
  bn_stats<<<1, C_, 0, stream>>>(S, gamma, beta, Acoef, Bcoef);

  const size_t total4 = (size_t)N_ * C_ * HW_ / 4;  // 4,194,304 float4
  bn_apply<<<(unsigned)(total4 / 256), 256, 0, stream>>>(x, Acoef, Bcoef, out);
}